// HypSupervisedContrastiveLoss_40054865002716
// MI455X (gfx1250) — compile-verified
//
#include <hip/hip_runtime.h>
#include <hip/hip_bf16.h>
#include <stdint.h>

typedef __attribute__((ext_vector_type(2))) float v2f;
typedef __attribute__((ext_vector_type(4))) float v4f;
typedef __attribute__((ext_vector_type(8))) float v8f;

#define B_N 4096
#define D_K 128
#define TILE 16
#define KSTEPS (D_K / 4)                       // 32 k-steps of V_WMMA_F32_16X16X4_F32
#define WAVES 8
#define COL_TILES (B_N / TILE)                 // 256
#define CSPLIT 4                               // column groups per row strip
#define TILES_PER_WG (COL_TILES / CSPLIT)      // 64
#define TILES_PER_WAVE (TILES_PER_WG / WAVES)  // 8

// Workspace layout (floats):
//   wsf[0] = loss_sum, wsf[1] = n_valid
//   sq = wsf + 2           (4096)
//   gS = wsf + 2 + 4096    (4096)  sum_{j!=i} exp(-2 dist) = sum q^20
//   gP = wsf + 2 + 8192    (4096)  sum_{positives} dist
//   gN = wsf + 2 + 12288   (4096)  num_pos
#define WS_SQ 2
#define WS_S (2 + B_N)
#define WS_P (2 + 2 * B_N)
#define WS_N (2 + 3 * B_N)

// ---------------------------------------------------------------------------
// Pass 1: squared row norms + zero all accumulators (ws is poisoned per run).
// ---------------------------------------------------------------------------
__global__ __launch_bounds__(256) void hyp_prep_kernel(const float* __restrict__ F,
                                                       float* __restrict__ wsf) {
    int i = blockIdx.x * blockDim.x + threadIdx.x;
    if (i == 0) { wsf[0] = 0.0f; wsf[1] = 0.0f; }
    if (i < B_N) {
        wsf[WS_S + i] = 0.0f;
        wsf[WS_P + i] = 0.0f;
        wsf[WS_N + i] = 0.0f;
        const v4f* p = (const v4f*)(F + (size_t)i * D_K);
        float s = 0.0f;
#pragma unroll
        for (int q = 0; q < D_K / 4; ++q) {
            v4f v = p[q];
            s += v[0] * v[0] + v[1] * v[1] + v[2] * v[2] + v[3] * v[3];
        }
        wsf[WS_SQ + i] = s;
    }
}

// ---------------------------------------------------------------------------
// Pass 2: fused Gram (f32 WMMA) + hyperbolic distance + row partials.
//
// Math: with s = sqrt(C*num_sq), ad = |den|:
//   q    = clamp_low( (ad - s) / (ad + s) )   == (1-nd)/(1+nd) with artanh clamp
//   dist = 20*artanh(nd) = -10*ln(q)
//   e    = exp(-dist/T)  = q^20   (5 multiplies, no v_exp)
//   logits_max == 0 analytically (diagonal distance is exactly 0).
//
// B fragments double-buffered in registers: next tile's 32 b64 loads issue in
// an s_clause and retire under partial s_wait_loadcnt during the math phase.
// ---------------------------------------------------------------------------
__global__ __launch_bounds__(256) void hyp_scl_kernel(const float* __restrict__ F,
                                                      const long long* __restrict__ labels,
                                                      float* __restrict__ wsf) {
    const float* __restrict__ sq = wsf + WS_SQ;
    __shared__ float rowS[TILE];
    __shared__ float rowP[TILE];
    __shared__ float rowN[TILE];

    const int tid    = threadIdx.x;
    const int lane   = tid & 31;
    const int wave   = tid >> 5;
    const int half   = lane >> 4;
    const int l16    = lane & 15;
    const int m_base = blockIdx.x * TILE;
    const int cgroup = blockIdx.y;

    if (tid < TILE) { rowS[tid] = 0.0f; rowP[tid] = 0.0f; rowN[tid] = 0.0f; }
    __syncthreads();

    // A fragments (16x4 f32 layout): lane L -> row L%16, k = 4*kk + 2*half + {0,1}.
    v2f afrag[KSTEPS];
    {
        const float* arow = F + (size_t)(m_base + l16) * D_K + 2 * half;
#pragma unroll
        for (int kk = 0; kk < KSTEPS; ++kk)
            afrag[kk] = *(const v2f*)(arow + 4 * kk);
    }

    float     sq_row[8];
    long long lab_row[8];
#pragma unroll
    for (int r = 0; r < 8; ++r) {
        int m = m_base + r + 8 * half;
        sq_row[r]  = sq[m];
        lab_row[r] = labels[m];
    }

    float accS[8], accP[8], accN[8];
#pragma unroll
    for (int r = 0; r < 8; ++r) { accS[r] = 0.0f; accP[r] = 0.0f; accN[r] = 0.0f; }

    const float Cc   = 0.01f;
    const float qmin = 1e-5f / (2.0f - 1e-5f);  // q at the artanh clamp

    const int ct0 = cgroup * TILES_PER_WG + wave;  // this wave's tiles: ct0 + t*WAVES

    // Prologue: load B fragments of the first tile into the register buffer.
    v2f bbuf[KSTEPS];
    {
        const float* b0 = F + (size_t)(ct0 * TILE + l16) * D_K + 2 * half;
#pragma unroll
        for (int kk = 0; kk < KSTEPS; ++kk)
            bbuf[kk] = *(const v2f*)(b0 + 4 * kk);
    }

    for (int t = 0; t < TILES_PER_WAVE; ++t) {
        const int ct = ct0 + t * WAVES;
        const int n  = ct * TILE + l16;

        const float     x2      = sq[n];
        const long long lab_col = labels[n];

        // Back-to-back WMMA chain over the staged fragments.
        v8f acc = {};
#pragma unroll
        for (int kk = 0; kk < KSTEPS; ++kk)
            acc = __builtin_amdgcn_wmma_f32_16x16x4_f32(
                false, afrag[kk], false, bbuf[kk], (short)0, acc, false, false);

        // Issue next tile's loads now; they complete during the math below.
        if (t + 1 < TILES_PER_WAVE) {
            const float* bn = F + (size_t)((ct + WAVES) * TILE + l16) * D_K + 2 * half;
#pragma unroll
            for (int kk = 0; kk < KSTEPS; ++kk)
                bbuf[kk] = *(const v2f*)(bn + 4 * kk);
        }

#pragma unroll
        for (int r = 0; r < 8; ++r) {
            const int   m   = m_base + r + 8 * half;
            const float G   = acc[r];
            const float xy  = -G;
            const float y2  = sq_row[r];
            const float aa  = 1.0f + 2.0f * Cc * xy + Cc * y2;
            const float bb  = 1.0f - Cc * x2;
            const float den = 1.0f + 2.0f * Cc * xy + Cc * Cc * x2 * y2;
            float num_sq = aa * aa * x2 + 2.0f * aa * bb * xy + bb * bb * y2;
            num_sq = fmaxf(num_sq, 0.0f);

            const float s  = sqrtf(Cc * num_sq);
            const float ad = fabsf(den);
            // q = (1-nd)/(1+nd) with nd = s/ad, clamped at the artanh limit.
            float q = (ad - s) / (ad + s);
            q = fmaxf(q, qmin);

            // e = exp(-2*dist) = q^20 via repeated squaring (no v_exp).
            const float q2  = q * q;
            const float q4  = q2 * q2;
            const float q8  = q4 * q4;
            const float q16 = q8 * q8;
            const float e   = q16 * q4;

            const bool diag = (m == n);
            accS[r] += diag ? 0.0f : e;

            const bool pos = (!diag) && (lab_row[r] == lab_col);
            const float dist = -10.0f * __logf(q);
            accP[r] += pos ? dist : 0.0f;
            accN[r] += pos ? 1.0f : 0.0f;
        }
    }

    // Reduce across the 16 lanes of each half.
#pragma unroll
    for (int r = 0; r < 8; ++r) {
#pragma unroll
        for (int msk = 8; msk >= 1; msk >>= 1) {
            accS[r] += __shfl_xor(accS[r], msk, 32);
            accP[r] += __shfl_xor(accP[r], msk, 32);
            accN[r] += __shfl_xor(accN[r], msk, 32);
        }
    }
    if (l16 == 0) {
#pragma unroll
        for (int r = 0; r < 8; ++r) {
            int row = r + 8 * half;
            atomicAdd(&rowS[row], accS[r]);
            atomicAdd(&rowP[row], accP[r]);
            atomicAdd(&rowN[row], accN[r]);
        }
    }
    __syncthreads();

    // One global atomic per row per workgroup into the per-row arrays.
    if (tid < TILE) {
        int row = m_base + tid;
        atomicAdd(&wsf[WS_S + row], rowS[tid]);
        atomicAdd(&wsf[WS_P + row], rowP[tid]);
        atomicAdd(&wsf[WS_N + row], rowN[tid]);
    }
}

// ---------------------------------------------------------------------------
// Pass 3: per-row loss (needs the completed global S sum) + scalar reduction.
// ---------------------------------------------------------------------------
__global__ __launch_bounds__(256) void hyp_rowloss_kernel(float* __restrict__ wsf) {
    int i = blockIdx.x * blockDim.x + threadIdx.x;
    if (i >= B_N) return;
    float S  = wsf[WS_S + i];
    float P  = wsf[WS_P + i];
    float np = wsf[WS_N + i];
    // row_loss = T * [ P/T + np*log(S) ] / max(np,1),  T = 0.5
    float row_loss = (P + 0.5f * np * __logf(S)) / fmaxf(np, 1.0f);
    float valid    = (np > 0.0f) ? 1.0f : 0.0f;
    if (isnan(row_loss)) row_loss = 0.0f;
    atomicAdd(&wsf[0], row_loss * valid);
    atomicAdd(&wsf[1], valid);
}

__global__ void hyp_finalize_kernel(const float* __restrict__ wsf,
                                    float* __restrict__ out) {
    out[0] = wsf[0] / fmaxf(wsf[1], 1.0f);
}

extern "C" void kernel_launch(void* const* d_in, const int* in_sizes, int n_in,
                              void* d_out, int out_size, void* d_ws, size_t ws_size,
                              hipStream_t stream) {
    const float*     F      = (const float*)d_in[0];
    const long long* labels = (const long long*)d_in[1];
    float*           wsf    = (float*)d_ws;
    float*           out    = (float*)d_out;

    hyp_prep_kernel<<<(B_N + 255) / 256, 256, 0, stream>>>(F, wsf);
    dim3 grid(COL_TILES, CSPLIT, 1);
    hyp_scl_kernel<<<grid, WAVES * 32, 0, stream>>>(F, labels, wsf);
    hyp_rowloss_kernel<<<(B_N + 255) / 256, 256, 0, stream>>>(wsf);
    hyp_finalize_kernel<<<1, 1, 0, stream>>>(wsf, out);
}